// OnnxScatterND_68367289418109
// MI455X (gfx1250) — compile-verified
//
#include <hip/hip_runtime.h>

// ---------------------------------------------------------------------------
// ScatterND-add: out = data; out[idx[i], :] += updates[i, :]
// data: 1,000,000 x 128 f32    updates: 500,000 x 128 f32    idx: 500,000 i32
//
// Pure HBM-bandwidth problem (AI ~0.04 FLOP/B; floor ~75us @ 23.3 TB/s).
// CDNA5 paths used:
//   - global_load/store_b128 with TH_NT (streaming copy, keep L2 for atomics)
//   - no-return GLOBAL_ATOMIC_ADD_F32 (hardware f32 add at L2, STOREcnt,
//     fire-and-forget)
//   - wave32 mapping: one wave per row, each atomic wave-op = one 128B line
//   - readfirstlane scalarization -> SGPR-base (saddr) addressing
// ---------------------------------------------------------------------------

typedef float f32x4 __attribute__((ext_vector_type(4)));

#define D 128          // row width (floats)
#define COPY_THREADS 256
#define SCAT_THREADS 256
#define WAVES_PER_BLOCK (SCAT_THREADS / 32)

__global__ void __launch_bounds__(COPY_THREADS)
scatternd_copy_kernel(const f32x4* __restrict__ src, f32x4* __restrict__ dst, int n4) {
    int i = blockIdx.x * COPY_THREADS + threadIdx.x;
    if (i < n4) {
        // NT temporal hint both ways: 1 GB streamed once, don't thrash L2.
        f32x4 v = __builtin_nontemporal_load(&src[i]);
        __builtin_nontemporal_store(v, &dst[i]);
    }
}

__global__ void __launch_bounds__(SCAT_THREADS)
scatternd_add_kernel(const int* __restrict__ indices,
                     const float* __restrict__ updates,
                     float* __restrict__ out, int n_upd) {
    const int lane = threadIdx.x & 31;

    // Wave-uniform row id, made explicit via readfirstlane so the compiler
    // keeps it (and everything derived from it) in SGPRs.
    int row = __builtin_amdgcn_readfirstlane(
        blockIdx.x * WAVES_PER_BLOCK + (threadIdx.x >> 5));
    if (row >= n_upd) return;

    // Scalar index load (uniform address) -> scalar base addresses.
    const int target = __builtin_amdgcn_readfirstlane(indices[row]);
    const float* __restrict__ src_row = updates + (long long)row    * D;
    float*       __restrict__ dst_row = out     + (long long)target * D;

    // lane l covers elements l, l+32, l+64, l+96:
    //  - each updates load: 32 lanes x 4B contiguous = one 128B line (NT:
    //    read-once data, preserve L2 for the atomic destinations)
    //  - each atomic wave-op: 32 consecutive 4B addrs = one 128B L2 line
    #pragma unroll
    for (int k = 0; k < 4; ++k) {
        const int off = lane + k * 32;
        float v = __builtin_nontemporal_load(&src_row[off]);
        // Discarded result -> no-return GLOBAL_ATOMIC_ADD_F32.
        unsafeAtomicAdd(&dst_row[off], v);
    }
}

extern "C" void kernel_launch(void* const* d_in, const int* in_sizes, int n_in,
                              void* d_out, int out_size, void* d_ws, size_t ws_size,
                              hipStream_t stream) {
    const float* data    = (const float*)d_in[0];   // 1,000,000 * 128 f32
    const int*   indices = (const int*)  d_in[1];   // 500,000 i32
    const float* updates = (const float*)d_in[2];   // 500,000 * 128 f32
    float*       out     = (float*)d_out;

    const int n_elems = in_sizes[0];      // 128,000,000
    const int n_upd   = in_sizes[1];      // 500,000

    // Phase 1: out = data (stream copy, 16B per thread)
    const int n4 = n_elems / 4;
    const int copy_blocks = (n4 + COPY_THREADS - 1) / COPY_THREADS;
    scatternd_copy_kernel<<<copy_blocks, COPY_THREADS, 0, stream>>>(
        (const f32x4*)data, (f32x4*)out, n4);

    // Phase 2: atomic scatter-add, one wave per update row (stream order
    // provides the copy->scatter dependency).
    const int scat_blocks = (n_upd + WAVES_PER_BLOCK - 1) / WAVES_PER_BLOCK;
    scatternd_add_kernel<<<scat_blocks, SCAT_THREADS, 0, stream>>>(
        indices, updates, out, n_upd);
}